// MeanSquaredError3D_66219805769835
// MI455X (gfx1250) — compile-verified
//
#include <hip/hip_runtime.h>
#include <math.h>

// ---- problem constants (from reference) ----
#define COL   14
#define PIX   (COL * COL)      // 196
#define NJ    24
#define TMPW  3
#define WAVES 8                // waves per block (wave32)
#define JPW   4                // joints per wave
#define JPB   (WAVES * JPW)    // 32 joints per block

#if __has_builtin(__builtin_amdgcn_global_load_async_to_lds_b128)
#define HAVE_ASYNC_LDS 1
#else
#define HAVE_ASYNC_LDS 0
#endif

// 16-byte GCC-style int vector, matching the builtin's parameter type
// ('__attribute__((__vector_size__(4 * sizeof(int)))) int *').
typedef int v4i __attribute__((vector_size(16)));
typedef __attribute__((address_space(1))) v4i* v4i_global_ptr;
typedef __attribute__((address_space(3))) v4i* v4i_lds_ptr;

__launch_bounds__(256)
__global__ void joint_loss_kernel(const float* __restrict__ o,
                                  const float* __restrict__ h,
                                  const float* __restrict__ t,
                                  const float* __restrict__ v,
                                  float* __restrict__ ws,
                                  int njoints)
{
    __shared__ __align__(16) float lds_h[JPB * PIX];   // 25088 B
    __shared__ float red_s[WAVES];
    __shared__ float red_c[WAVES];

    const int tid   = threadIdx.x;
    const int lane  = tid & 31;
    const int wave  = tid >> 5;
    const int jbase = blockIdx.x * JPB;
    const int jcnt  = (njoints - jbase < JPB) ? (njoints - jbase) : JPB;

    // ---- stage this block's contiguous heatmap slab into LDS ----
    const float* hblk  = h + (size_t)jbase * PIX;
    const int    nlines = (jcnt * PIX) >> 2;           // float4 lines (196*jcnt % 4 == 0)
    for (int L = tid; L < nlines; L += 256) {
#if HAVE_ASYNC_LDS
        __builtin_amdgcn_global_load_async_to_lds_b128(
            (v4i_global_ptr)(void*)(hblk + (size_t)L * 4),
            (v4i_lds_ptr)(void*)(&lds_h[L * 4]),
            0, 0);
#else
        *(float4*)(&lds_h[L * 4]) = *(const float4*)(hblk + (size_t)L * 4);
#endif
        // speculative stream-prefetch of the next block's matching line
        __builtin_prefetch(hblk + (size_t)JPB * PIX + (size_t)L * 4, 0, 3);
    }
#if HAVE_ASYNC_LDS
    asm volatile("s_wait_asynccnt 0" ::: "memory");
#endif
    __syncthreads();

    float waveS = 0.0f;   // valid on lane 0
    float waveC = 0.0f;

    for (int q = 0; q < JPW; ++q) {
        const int jl = wave * JPW + q;      // joint index within block
        const int g  = jbase + jl;          // global joint index
        if (g >= njoints) break;
        const int b = g / NJ;
        const int j = g - b * NJ;
        const float* __restrict__ hj = &lds_h[jl * PIX];

        // ---- argmax (first-occurrence tie-break, as jnp.argmax) ----
        float hv[7];
        float bestV = -INFINITY;
        int   bestI = 0x7fffffff;
#pragma unroll
        for (int k = 0; k < 7; ++k) {
            const int idx = lane + 32 * k;
            const float val = (idx < PIX) ? hj[idx] : -INFINITY;
            hv[k] = val;
            if (val > bestV) { bestV = val; bestI = idx; }   // strided: k↑ ⇒ idx↑
        }
#pragma unroll
        for (int off = 16; off; off >>= 1) {
            const float ov = __shfl_xor(bestV, off, 32);
            const int   oi = __shfl_xor(bestI, off, 32);
            if (ov > bestV || (ov == bestV && oi < bestI)) { bestV = ov; bestI = oi; }
        }
        const int yC = bestI / COL;
        const int xC = bestI - yC * COL;

        // ---- target / visibility (broadcast loads) ----
        const float t0 = t[(size_t)g * 3 + 0];
        const float t1 = t[(size_t)g * 3 + 1];
        const float t2 = t[(size_t)g * 3 + 2];
        const float v0 = v[(size_t)g * 3 + 0];

        const int mu_x = (int)truncf(t0 * (float)COL + 0.5f);
        const int mu_y = (int)truncf(t1 * (float)COL + 0.5f);
        const bool valid = (mu_x - TMPW < COL) && (mu_x + TMPW + 1 >= 0) &&
                           (mu_y - TMPW < COL) && (mu_y + TMPW + 1 >= 0);
        const bool vis   = ((int)v0 == 1);
        const bool stamp = vis && valid;    // == mask1; cnt == N2

        if (stamp) {
            // ---- heatmap MSE vs separable 7x7 gaussian ----
            float s1 = 0.0f;
#pragma unroll
            for (int k = 0; k < 7; ++k) {
                const int idx = lane + 32 * k;
                if (idx < PIX) {
                    const int py = idx / COL;
                    const int px = idx - py * COL;
                    const int dx = px - mu_x;
                    const int dy = py - mu_y;
                    float tt = 0.0f;
                    if (dx >= -TMPW && dx <= TMPW && dy >= -TMPW && dy <= TMPW)
                        tt = expf(-0.5f * (float)(dx * dx + dy * dy));
                    const float d = hv[k] - tt;
                    s1 += d * d;
                }
            }
#pragma unroll
            for (int off = 16; off; off >>= 1)
                s1 += __shfl_xor(s1, off, 32);

            if (lane == 0) {
                // ---- 3-point gather from o at argmax location ----
                const size_t obase = ((size_t)b * 3 * NJ + j) * PIX
                                   + (size_t)yC * COL + xC;
                const float ox = o[obase];
                const float oy = o[obase + (size_t)NJ * PIX];
                const float oz = o[obase + (size_t)2 * NJ * PIX];
                const float scale = 1.0f / (float)COL;
                const float d0 = (ox + (float)xC * scale) - t0;
                const float d1 = (oy + (float)yC * scale) - t1;
                const float d2 = oz - t2;
                waveS += s1 + d0 * d0 + d1 * d1 + d2 * d2;
                waveC += 1.0f;
            }
        }
    }

    if (lane == 0) { red_s[wave] = waveS; red_c[wave] = waveC; }
    __syncthreads();
    if (tid == 0) {
        float S = 0.0f, C = 0.0f;
#pragma unroll
        for (int w = 0; w < WAVES; ++w) { S += red_s[w]; C += red_c[w]; }
        ws[2 * blockIdx.x + 0] = S;
        ws[2 * blockIdx.x + 1] = C;
    }
}

__launch_bounds__(256)
__global__ void finalize_kernel(const float* __restrict__ ws,
                                float* __restrict__ out,
                                int nblocks)
{
    __shared__ float ss[256];
    __shared__ float cc[256];
    const int tid = threadIdx.x;
    float S = 0.0f, C = 0.0f;
    for (int i = tid; i < nblocks; i += 256) {
        S += ws[2 * i + 0];
        C += ws[2 * i + 1];
    }
    ss[tid] = S; cc[tid] = C;
    __syncthreads();
    for (int st = 128; st; st >>= 1) {
        if (tid < st) { ss[tid] += ss[tid + st]; cc[tid] += cc[tid + st]; }
        __syncthreads();
    }
    if (tid == 0) out[0] = ss[0] / cc[0];   // d1 + d2 == (S1+S2)/cnt since cnt==N2
}

extern "C" void kernel_launch(void* const* d_in, const int* in_sizes, int n_in,
                              void* d_out, int out_size, void* d_ws, size_t ws_size,
                              hipStream_t stream)
{
    const float* o = (const float*)d_in[0];
    const float* h = (const float*)d_in[1];
    const float* t = (const float*)d_in[2];
    const float* v = (const float*)d_in[3];

    const int njoints = in_sizes[2] / 3;                 // B * Nj
    const int nblocks = (njoints + JPB - 1) / JPB;       // 3072 for B=4096,Nj=24

    joint_loss_kernel<<<nblocks, 256, 0, stream>>>(o, h, t, v, (float*)d_ws, njoints);
    finalize_kernel<<<1, 256, 0, stream>>>((const float*)d_ws, (float*)d_out, nblocks);
}